// TopKOnlyCoordination_84593675862308
// MI455X (gfx1250) — compile-verified
//
#include <hip/hip_runtime.h>

typedef __attribute__((ext_vector_type(2))) float v2f;
typedef __attribute__((ext_vector_type(8))) float v8f;

#define NTOK 256
#define DDIM 128
#define HDIM 256
#define PSTR 132   // padded LDS row stride (floats) -> conflict-free A fragments

// ---------- kernel 1: separable projections  ab1 = s@W1a + b1,  ab2 = s@W1b ----------
__global__ __launch_bounds__(256) void proj_kernel(
    const float* __restrict__ s, const float* __restrict__ W1,
    const float* __restrict__ b1, float* __restrict__ ab1, float* __restrict__ ab2)
{
  __shared__ float srow[DDIM];
  int tid = threadIdx.x;            // h = 0..255
  int row = blockIdx.x;             // b*256 + n
  if (tid < DDIM) srow[tid] = s[(size_t)row * DDIM + tid];
  __syncthreads();
  float a1 = 0.f, a2 = 0.f;
#pragma unroll 8
  for (int d = 0; d < DDIM; ++d) {
    float sv = srow[d];
    a1 = fmaf(sv, W1[(size_t)d * HDIM + tid], a1);
    a2 = fmaf(sv, W1[(size_t)(DDIM + d) * HDIM + tid], a2);
  }
  ab1[(size_t)row * HDIM + tid] = a1 + b1[tid];   // fold b1 into the i-term
  ab2[(size_t)row * HDIM + tid] = a2;
}

// ---------- kernel 2: repack W1c (rows 256..383 of W1) into k-pair-major ----------
// w1cp[pair][h][2] with pair = d/2, so a lane's B fragment (k,k+1 at fixed h)
// is one contiguous 8-byte global load.
__global__ __launch_bounds__(256) void pack_kernel(const float* __restrict__ W1,
                                                   float* __restrict__ w1cp)
{
  int d = blockIdx.x;   // 0..127
  int h = threadIdx.x;  // 0..255
  w1cp[(size_t)(d >> 1) * (HDIM * 2) + h * 2 + (d & 1)] =
      W1[(size_t)(2 * DDIM + d) * HDIM + h];
}

// ---------- kernel 3: pairwise GEMM (WMMA f32) + fused score/topk/gather ----------
__global__ __launch_bounds__(256) void pair_topk_kernel(
    const float* __restrict__ s, const float* __restrict__ W2,
    const float* __restrict__ b2, const float* __restrict__ ab1,
    const float* __restrict__ ab2, const float* __restrict__ w1cp,
    float* __restrict__ out)
{
  __shared__ float sP[64 * PSTR];   // P chunk: 64 j-rows x 128 d (padded)
  __shared__ float sI[DDIM];        // s[b,i,:]
  __shared__ float sAi[HDIM];       // ab1[b,i,:]  (b1 folded)
  __shared__ float sW2[HDIM];
  __shared__ float sScore[NTOK];
  __shared__ int   sFlag[NTOK];
  __shared__ int   sSel[8];

  const int tid  = threadIdx.x;
  const int wg   = blockIdx.x;          // b*256 + i
  const int b    = wg >> 8;
  const int i    = wg & 255;
  const int wave = tid >> 5;
  const int lane = tid & 31;
  const int hi   = lane >> 4;           // half-wave select
  const int l15  = lane & 15;

  const float* sb = s + (size_t)b * NTOK * DDIM;

  if (tid < DDIM) sI[tid] = sb[(size_t)i * DDIM + tid];
  sAi[tid]    = ab1[(size_t)wg * HDIM + tid];
  sW2[tid]    = W2[tid];
  sScore[tid] = b2[0];
  sFlag[tid]  = 0;
  __syncthreads();

  for (int chunk = 0; chunk < 4; ++chunk) {
    const int j0 = chunk * 64;
    // ---- build P[j,d] = s_i[d] * s_j[d] for 64 rows ----
    {
      int row   = tid >> 2;          // 0..63
      int dbase = (tid & 3) * 32;    // 4 threads per row, 32 d each
      const float* src = sb + (size_t)(j0 + row) * DDIM + dbase;
      float* dst = sP + row * PSTR + dbase;
#pragma unroll
      for (int q = 0; q < 32; q += 4) {
        float4 v = *(const float4*)(src + q);
        dst[q + 0] = v.x * sI[dbase + q + 0];
        dst[q + 1] = v.y * sI[dbase + q + 1];
        dst[q + 2] = v.z * sI[dbase + q + 2];
        dst[q + 3] = v.w * sI[dbase + q + 3];
      }
    }
    __syncthreads();

    // ---- 8 tiles per wave: 4 j-subtiles (16 each) x 2 h-tiles ----
    for (int t = 0; t < 8; ++t) {
      const int jb = (t >> 1) * 16;                // j offset within chunk
      const int hb = (wave * 2 + (t & 1)) * 16;    // h tile base (0..255)

      v8f acc = {0.f, 0.f, 0.f, 0.f, 0.f, 0.f, 0.f, 0.f};
      const float* pA = sP + (jb + l15) * PSTR + 2 * hi;
      const float* pB = w1cp + ((size_t)hi * HDIM + (hb + l15)) * 2;
#pragma unroll
      for (int ks = 0; ks < 32; ++ks) {            // K = 128, 4 per WMMA
        v2f a  = *(const v2f*)(pA + ks * 4);
        v2f bb = *(const v2f*)(pB + (size_t)ks * 2 * HDIM * 2);
        acc = __builtin_amdgcn_wmma_f32_16x16x4_f32(
            false, a, false, bb, (short)0, acc, false, false);
      }

      // ---- fused epilogue: relu(C + Ai + Bv) * W2, reduce over h ----
      const int   h    = hb + l15;
      const float aih  = sAi[h];
      const float w2h  = sW2[h];
      const float* bvr = ab2 + ((size_t)(b * NTOK + j0 + jb + 8 * hi)) * HDIM + h;
#pragma unroll
      for (int r = 0; r < 8; ++r) {
        int   j = j0 + jb + r + 8 * hi;            // global j for this element
        float v = acc[r] + aih + bvr[(size_t)r * HDIM];
        v = fmaxf(v, 0.f) * w2h;
#pragma unroll
        for (int off = 8; off >= 1; off >>= 1)      // sum 16 h-lanes
          v += __shfl_xor(v, off, 16);
        if (l15 == 0) atomicAdd(&sScore[j], v);     // ds_add_f32
      }
    }
    __syncthreads();
  }

  // ---- top-8 per row (wave 0): iterative argmax, ties -> lowest index ----
  if (tid < 32) {
    volatile float* vs = sScore;
    for (int kk = 0; kk < 8; ++kk) {
      float best = -3.0e38f;
      int   bidx = 0x7fffffff;
      for (int j = lane; j < NTOK; j += 32) {
        float v = vs[j];
        if (v > best) { best = v; bidx = j; }
      }
#pragma unroll
      for (int off = 16; off >= 1; off >>= 1) {
        float ov = __shfl_xor(best, off, 32);
        int   oi = __shfl_xor(bidx, off, 32);
        if (ov > best || (ov == best && oi < bidx)) { best = ov; bidx = oi; }
      }
      if (lane == 0) { sFlag[bidx] = 1; sSel[kk] = bidx; vs[bidx] = -3.0e38f; }
      // in-wave DS ordering: lane-0 write precedes next iteration's reads
    }
  }
  __syncthreads();

  // ---- outputs: ctx [B,N,D], gate [B,N,N], w [B,N,N] (concat, flat) ----
  float* gate_out = out + (size_t)4 * NTOK * DDIM;
  float* w_out    = gate_out + (size_t)4 * NTOK * NTOK;
  float g = sFlag[tid] ? 1.0f : 0.0f;
  gate_out[(size_t)wg * NTOK + tid] = g;
  w_out[(size_t)wg * NTOK + tid]    = g * 0.125f;     // sum(gate)=8 exactly
  if (tid < DDIM) {
    float a = 0.f;
#pragma unroll
    for (int kk = 0; kk < 8; ++kk)
      a += sb[(size_t)sSel[kk] * DDIM + tid];
    out[(size_t)wg * DDIM + tid] = a * 0.125f;
  }
}

extern "C" void kernel_launch(void* const* d_in, const int* in_sizes, int n_in,
                              void* d_out, int out_size, void* d_ws, size_t ws_size,
                              hipStream_t stream) {
  (void)in_sizes; (void)n_in; (void)out_size; (void)ws_size;
  const float* s  = (const float*)d_in[0];
  const float* W1 = (const float*)d_in[1];
  const float* b1 = (const float*)d_in[2];
  const float* W2 = (const float*)d_in[3];
  const float* b2 = (const float*)d_in[4];
  // d_in[5] = K (==8, fixed by setup)

  float* ab1  = (float*)d_ws;                    // [1024, 256]
  float* ab2  = ab1 + (size_t)1024 * HDIM;       // [1024, 256]
  float* w1cp = ab2 + (size_t)1024 * HDIM;       // [64, 256, 2] k-pair-major W1c

  proj_kernel<<<1024, 256, 0, stream>>>(s, W1, b1, ab1, ab2);
  pack_kernel<<<128, 256, 0, stream>>>(W1, w1cp);
  pair_topk_kernel<<<1024, 256, 0, stream>>>(s, W2, b2, ab1, ab2, w1cp,
                                             (float*)d_out);
}